// Network_17351667875938
// MI455X (gfx1250) — compile-verified
//
#include <hip/hip_runtime.h>
#include <hip/hip_bf16.h>
#include <math.h>

typedef __attribute__((ext_vector_type(2))) float v2f;
typedef __attribute__((ext_vector_type(8))) float v8f;

#define NNODES 20000
#define NEDGES 160000
#define NREL 4
#define NHEAD 4
#define DCH 128
#define HD 512
#define NB 200
#define NT 8
#define NA 10
#define NCOSF 64
#define BT (NB*NT)

__device__ __forceinline__ float selu_f(float v) {
    const float SC = 1.0507009873554805f, AL = 1.6732632423543772f;
    return v > 0.f ? SC * v : SC * AL * (expf(v) - 1.f);
}

// ---------------- CSR construction ----------------
__global__ void k_zero_i32(int* p, int n) {
    int i = blockIdx.x * blockDim.x + threadIdx.x;
    if (i < n) p[i] = 0;
}

__global__ void k_count(const int* dst, int* deg, int e) {
    int i = blockIdx.x * blockDim.x + threadIdx.x;
    if (i < e) atomicAdd(&deg[dst[i]], 1);
}

__global__ void k_scan(const int* deg, int* rowptr, int n) {
    __shared__ int buf[1024];
    __shared__ int carry_s;
    int t = threadIdx.x;
    if (t == 0) carry_s = 0;
    __syncthreads();
    for (int base = 0; base < n; base += 1024) {
        int i = base + t;
        buf[t] = (i < n) ? deg[i] : 0;
        __syncthreads();
        for (int off2 = 1; off2 < 1024; off2 <<= 1) {
            int add = (t >= off2) ? buf[t - off2] : 0;
            __syncthreads();
            buf[t] += add;
            __syncthreads();
        }
        int carry = carry_s;
        if (i < n) rowptr[i + 1] = carry + buf[t];
        __syncthreads();
        if (t == 1023) carry_s = carry + buf[1023];
        __syncthreads();
    }
    if (t == 0) rowptr[0] = 0;
}

__global__ void k_copy_i32(const int* a, int* b, int n) {
    int i = blockIdx.x * blockDim.x + threadIdx.x;
    if (i < n) b[i] = a[i];
}

__global__ void k_fill(const int* dst, int* cursor, int* eids, int e) {
    int i = blockIdx.x * blockDim.x + threadIdx.x;
    if (i < e) {
        int pos = atomicAdd(&cursor[dst[i]], 1);
        eids[pos] = i;
    }
}

// per-node insertion sort -> deterministic accumulation order
__global__ void k_sortnode(const int* rowptr, int* eids, int n) {
    int i = blockIdx.x * blockDim.x + threadIdx.x;
    if (i >= n) return;
    int b = rowptr[i], e = rowptr[i + 1];
    for (int j = b + 1; j < e; ++j) {
        int v = eids[j];
        int p = j - 1;
        while (p >= b && eids[p] > v) { eids[p + 1] = eids[p]; --p; }
        eids[p + 1] = v;
    }
}

// ---------------- layer-1 alpha projection tables ----------------
// Pq[r,i,h] = sum_c W1[r,i,c]*q1[c,h]  (and Pk with k1)
__global__ void k_prep_p1(const float* W1, const float* q1, const float* k1,
                          float* Pq, float* Pk) {
    int j = threadIdx.x; // 64 threads
    const float* qk = (j < 32) ? q1 : k1;
    float* P = (j < 32) ? Pq : Pk;
    int jj = j & 31;
    int r = jj >> 3, ii = (jj >> 2) & 1, h = jj & 3;
    const float* w = W1 + (r * 2 + ii) * HD;
    float s = 0.f;
    for (int c = 0; c < HD; ++c) s += w[c] * qk[c * NHEAD + h];
    P[jj] = s;
}

// transposed padded [q|k]: qkpT[j][kk] (16 x 512, row-major) so GEMM B loads are contiguous
__global__ void k_qkpadT(const float* q, const float* k, float* qkpT) {
    int i = blockIdx.x * blockDim.x + threadIdx.x;
    if (i >= 16 * HD) return;
    int j = i / HD, kk = i % HD;
    float v = 0.f;
    if (j < 4) v = q[kk * NHEAD + j];
    else if (j < 8) v = k[kk * NHEAD + (j - 4)];
    qkpT[i] = v;
}

// W2 [r][k][o] -> W2T [n=r*512+o][k]  (2048 x 128, row-major)
__global__ void k_w2t(const float* W2, float* W2T) {
    int i = blockIdx.x * blockDim.x + threadIdx.x;
    if (i >= NREL * DCH * HD) return;
    int r = i / (DCH * HD);
    int k = (i / HD) % DCH;
    int o = i % HD;
    W2T[(size_t)(r * HD + o) * DCH + k] = W2[i];
}

__global__ void k_noisy(const float* w, const float* s, const float* e, float* o, int n) {
    int i = blockIdx.x * blockDim.x + threadIdx.x;
    if (i < n) o[i] = w[i] + s[i] * e[i];
}

// ---------------- WMMA f32 GEMM ----------------
// C[M,N] = A[M,K] * B^T where B is [N,K] row-major (ldb).  A tile (16 x KT) staged in LDS,
// one wave per 16x16 C tile, 4 waves (64 N-columns) per block share the A tile.
// ACT: 0 = plain store, 1 = selu(val + bias[n])
template<int KT, int ACT>
__global__ __launch_bounds__(128) void k_wmma_gemm(
        const float* __restrict__ A, const float* __restrict__ B,
        float* __restrict__ C, int M, int N,
        int lda, int ldb, int ldc, const float* __restrict__ bias) {
    __shared__ float sA[16 * KT];
    int tid = threadIdx.x;
    int lane = tid & 31, wave = tid >> 5;
    int m0 = blockIdx.y * 16;
    int n0 = blockIdx.x * 64 + wave * 16;
    // cooperative stage of A tile (all threads; barrier is block-uniform)
    for (int i = tid; i < 16 * KT; i += 128) {
        int r = i / KT, ccol = i % KT;  // KT is a power of two -> shifts
        sA[i] = A[(size_t)(m0 + r) * lda + ccol];
    }
    __syncthreads();
    if (n0 < N) {                       // wave-uniform; EXEC stays all-ones for WMMA
        int l16 = lane & 15;
        int khalf = (lane >> 4) * 2;    // lanes 0-15 -> K {0,1}, lanes 16-31 -> K {2,3}
        int ncol = n0 + l16;
        const float* sArow = sA + l16 * KT + khalf;
        const float* Brow = B + (size_t)ncol * ldb + khalf;
        v8f c = {};
        for (int k0 = 0; k0 < KT; k0 += 4) {
            v2f a, b;
            a.x = sArow[k0];
            a.y = sArow[k0 + 1];
            b.x = Brow[k0];
            b.y = Brow[k0 + 1];
            c = __builtin_amdgcn_wmma_f32_16x16x4_f32(false, a, false, b, (short)0, c,
                                                      false, false);
        }
        int rbase = m0 + 8 * (lane >> 4);
        for (int v = 0; v < 8; ++v) {
            float val = c[v];
            if (ACT == 1) val = selu_f(val + bias[ncol]);
            C[(size_t)(rbase + v) * ldc + ncol] = val;
        }
    }
}

// ---------------- RGAT layer 1: fused alpha/softmax/aggregate, xw1 on the fly ----------------
__global__ void k_agg1(const int* rowptr, const int* eids, const int* srcv, const int* etype,
                       const float* x, const float* W1, const float* Pq, const float* Pk,
                       const float* bias, float* hout) {
    int n = blockIdx.x, t = threadIdx.x; // 128 threads
    __shared__ float sW1[NREL * 2 * HD];   // 16 KB
    __shared__ float sPq[32], sPk[32];
    __shared__ float s_red[128];
    __shared__ float s_w[32 * 4];
    __shared__ float s_x0[32], s_x1[32];
    __shared__ int s_et[32];
    __shared__ float s_max[4], s_inv[4];
    for (int i = t; i < NREL * 2 * HD; i += 128) sW1[i] = W1[i];
    if (t < 32) { sPq[t] = Pq[t]; sPk[t] = Pk[t]; }
    __syncthreads();
    int beg = rowptr[n], cnt = rowptr[n + 1] - beg;
    int h = t & 3;
    float xn0 = x[n * 2], xn1 = x[n * 2 + 1];
    float mx = -3.4e38f;
    for (int idx = t; idx < cnt * 4; idx += 128) {
        int i = idx >> 2;
        int e = eids[beg + i];
        int s = srcv[e], et = etype[e];
        float a = xn0 * sPq[et * 8 + h] + xn1 * sPq[et * 8 + 4 + h]
                + x[s * 2] * sPk[et * 8 + h] + x[s * 2 + 1] * sPk[et * 8 + 4 + h];
        a = a > 0.f ? a : 0.2f * a;
        mx = fmaxf(mx, a);
    }
    s_red[t] = mx;
    __syncthreads();
    for (int off2 = 64; off2 >= 4; off2 >>= 1) {
        if (t < off2) s_red[t] = fmaxf(s_red[t], s_red[t + off2]);
        __syncthreads();
    }
    if (t < 4) s_max[t] = s_red[t];
    __syncthreads();
    float mxh = s_max[h], sm = 0.f;
    for (int idx = t; idx < cnt * 4; idx += 128) {
        int i = idx >> 2;
        int e = eids[beg + i];
        int s = srcv[e], et = etype[e];
        float a = xn0 * sPq[et * 8 + h] + xn1 * sPq[et * 8 + 4 + h]
                + x[s * 2] * sPk[et * 8 + h] + x[s * 2 + 1] * sPk[et * 8 + 4 + h];
        a = a > 0.f ? a : 0.2f * a;
        sm += expf(a - mxh);
    }
    s_red[t] = sm;
    __syncthreads();
    for (int off2 = 64; off2 >= 4; off2 >>= 1) {
        if (t < off2) s_red[t] += s_red[t + off2];
        __syncthreads();
    }
    if (t < 4) s_inv[t] = (cnt > 0) ? 1.f / s_red[t] : 0.f;
    __syncthreads();
    float acc0 = 0.f, acc1 = 0.f, acc2 = 0.f, acc3 = 0.f;
    for (int cb = 0; cb < cnt; cb += 32) {
        int csz = min(32, cnt - cb);
        if (t < csz * 4) {
            int i = t >> 2, hh = t & 3;
            int e = eids[beg + cb + i];
            int s = srcv[e], et = etype[e];
            float xs0 = x[s * 2], xs1 = x[s * 2 + 1];
            if (hh == 0) { s_x0[i] = xs0; s_x1[i] = xs1; s_et[i] = et; }
            float a = xn0 * sPq[et * 8 + hh] + xn1 * sPq[et * 8 + 4 + hh]
                    + xs0 * sPk[et * 8 + hh] + xs1 * sPk[et * 8 + 4 + hh];
            a = a > 0.f ? a : 0.2f * a;
            s_w[t] = expf(a - s_max[hh]) * s_inv[hh];
        }
        __syncthreads();
        for (int i = 0; i < csz; ++i) {
            int et = s_et[i];
            float x0 = s_x0[i], x1 = s_x1[i];
            const float* w0 = sW1 + et * (2 * HD);
            acc0 += s_w[i * 4 + 0] * (x0 * w0[t] + x1 * w0[HD + t]);
            acc1 += s_w[i * 4 + 1] * (x0 * w0[128 + t] + x1 * w0[HD + 128 + t]);
            acc2 += s_w[i * 4 + 2] * (x0 * w0[256 + t] + x1 * w0[HD + 256 + t]);
            acc3 += s_w[i * 4 + 3] * (x0 * w0[384 + t] + x1 * w0[HD + 384 + t]);
        }
        __syncthreads();
    }
    hout[(size_t)n * DCH + t] = selu_f(0.25f * (acc0 + acc1 + acc2 + acc3) + bias[t]);
}

// ---------------- RGAT layer 2: fused softmax/aggregate from xw2 + aiaj tables ----------------
__global__ void k_agg2(const int* rowptr, const int* eids, const int* srcv, const int* etype,
                       const float* aiaj, const float* xw, const float* bias, float* hout) {
    int n = blockIdx.x, t = threadIdx.x; // 128
    __shared__ float s_red[128];
    __shared__ float s_w[32 * 4];
    __shared__ int s_src[32];
    __shared__ int s_et[32];
    __shared__ float s_max[4], s_inv[4];
    int beg = rowptr[n], cnt = rowptr[n + 1] - beg;
    int h = t & 3;
    float mx = -3.4e38f;
    for (int idx = t; idx < cnt * 4; idx += 128) {
        int i = idx >> 2;
        int e = eids[beg + i];
        int s = srcv[e], et = etype[e];
        float a = aiaj[(size_t)(n * NREL + et) * 16 + h] + aiaj[(size_t)(s * NREL + et) * 16 + 4 + h];
        a = a > 0.f ? a : 0.2f * a;
        mx = fmaxf(mx, a);
    }
    s_red[t] = mx;
    __syncthreads();
    for (int off2 = 64; off2 >= 4; off2 >>= 1) {
        if (t < off2) s_red[t] = fmaxf(s_red[t], s_red[t + off2]);
        __syncthreads();
    }
    if (t < 4) s_max[t] = s_red[t];
    __syncthreads();
    float mxh = s_max[h], sm = 0.f;
    for (int idx = t; idx < cnt * 4; idx += 128) {
        int i = idx >> 2;
        int e = eids[beg + i];
        int s = srcv[e], et = etype[e];
        float a = aiaj[(size_t)(n * NREL + et) * 16 + h] + aiaj[(size_t)(s * NREL + et) * 16 + 4 + h];
        a = a > 0.f ? a : 0.2f * a;
        sm += expf(a - mxh);
    }
    s_red[t] = sm;
    __syncthreads();
    for (int off2 = 64; off2 >= 4; off2 >>= 1) {
        if (t < off2) s_red[t] += s_red[t + off2];
        __syncthreads();
    }
    if (t < 4) s_inv[t] = (cnt > 0) ? 1.f / s_red[t] : 0.f;
    __syncthreads();
    float acc0 = 0.f, acc1 = 0.f, acc2 = 0.f, acc3 = 0.f;
    for (int cb = 0; cb < cnt; cb += 32) {
        int csz = min(32, cnt - cb);
        if (t < csz * 4) {
            int i = t >> 2, hh = t & 3;
            int e = eids[beg + cb + i];
            int s = srcv[e], et = etype[e];
            if (hh == 0) { s_src[i] = s; s_et[i] = et; }
            float a = aiaj[(size_t)(n * NREL + et) * 16 + hh] + aiaj[(size_t)(s * NREL + et) * 16 + 4 + hh];
            a = a > 0.f ? a : 0.2f * a;
            s_w[t] = expf(a - s_max[hh]) * s_inv[hh];
        }
        __syncthreads();
        for (int i = 0; i < csz; ++i) {
            const float* row = xw + (size_t)(s_src[i] * NREL + s_et[i]) * HD;
            acc0 += s_w[i * 4 + 0] * row[t];
            acc1 += s_w[i * 4 + 1] * row[128 + t];
            acc2 += s_w[i * 4 + 2] * row[256 + t];
            acc3 += s_w[i * 4 + 3] * row[384 + t];
        }
        __syncthreads();
    }
    hout[(size_t)n * DCH + t] = selu_f(0.25f * (acc0 + acc1 + acc2 + acc3) + bias[t]);
}

// ---------------- pooling / cosine embedding / head ----------------
__global__ void k_pool(const float* h, float* g) {
    int b = blockIdx.x, t = threadIdx.x;
    const float* base = h + (size_t)b * 100 * DCH;
    float s = 0.f;
    for (int i = 0; i < 100; ++i) s += base[i * DCH + t];
    g[b * DCH + t] = s;
}

__global__ void k_cosz(const float* taus, const float* cosw, const float* cosb,
                       const float* g, float* z) {
    int row = blockIdx.x; // b*T + t
    int d = threadIdx.x;
    int b = row >> 3;
    float tau = taus[row];
    float acc = 0.f;
    for (int i = 0; i < NCOSF; ++i)
        acc += cosf(tau * (3.14159265358979323846f * (float)(i + 1))) * cosw[d * NCOSF + i];
    acc += cosb[d];
    acc = acc > 0.f ? acc : 0.f;
    z[(size_t)row * DCH + d] = g[b * DCH + d] * acc;
}

__global__ void k_head(const float* y2, const float* adveff, const float* advbe,
                       const float* valeff, const float* valbe, float* out) {
    int m = blockIdx.x, t = threadIdx.x; // 32 threads
    __shared__ float s_adv[NA];
    __shared__ float s_val;
    const float* zr = y2 + (size_t)m * DCH;
    if (t < NA) {
        float a = advbe[t];
        for (int k = 0; k < DCH; ++k) a += zr[k] * adveff[t * DCH + k];
        s_adv[t] = a;
    }
    if (t == NA) {
        float v = valbe[0];
        for (int k = 0; k < DCH; ++k) v += zr[k] * valeff[k];
        s_val = v;
    }
    __syncthreads();
    if (t == 0) {
        float mean = 0.f;
        for (int a = 0; a < NA; ++a) mean += s_adv[a];
        mean *= (1.f / NA);
        for (int a = 0; a < NA; ++a) out[m * NA + a] = s_val + s_adv[a] - mean;
    }
}

__global__ void k_copy_f32(const float* a, float* b, int n) {
    int i = blockIdx.x * blockDim.x + threadIdx.x;
    if (i < n) b[i] = a[i];
}

extern "C" void kernel_launch(void* const* d_in, const int* in_sizes, int n_in,
                              void* d_out, int out_size, void* d_ws, size_t ws_size,
                              hipStream_t stream) {
    const float* x       = (const float*)d_in[0];
    const int*   eidx    = (const int*)d_in[1];
    const int*   etype   = (const int*)d_in[2];
    const float* taus    = (const float*)d_in[3];
    const float* W1      = (const float*)d_in[4];
    const float* q1      = (const float*)d_in[5];
    const float* k1      = (const float*)d_in[6];
    const float* b1      = (const float*)d_in[7];
    const float* W2      = (const float*)d_in[8];
    const float* q2      = (const float*)d_in[9];
    const float* k2      = (const float*)d_in[10];
    const float* b2      = (const float*)d_in[11];
    const float* cosw    = (const float*)d_in[12];
    const float* cosb    = (const float*)d_in[13];
    float* out = (float*)d_out;

    const int* srcv = eidx;
    const int* dstv = eidx + NEDGES;

    char* w = (char*)d_ws;
    size_t off = 0;
    auto grab = [&](size_t bytes) -> void* {
        void* p = w + off;
        off += (bytes + 255) & ~(size_t)255;
        return p;
    };
    float* xw2    = (float*)grab((size_t)NNODES * NREL * HD * 4); // 164 MB, L2-resident
    float* aiaj   = (float*)grab((size_t)NNODES * NREL * 16 * 4);
    float* h1     = (float*)grab((size_t)NNODES * DCH * 4);
    float* h2     = (float*)grab((size_t)NNODES * DCH * 4);
    int*   deg    = (int*)grab(NNODES * 4);
    int*   rowptr = (int*)grab((NNODES + 1) * 4);
    int*   cursor = (int*)grab(NNODES * 4);
    int*   eids   = (int*)grab(NEDGES * 4);
    float* Pq     = (float*)grab(32 * 4);
    float* Pk     = (float*)grab(32 * 4);
    float* qkpT   = (float*)grab(16 * HD * 4);
    float* W2T    = (float*)grab((size_t)NREL * DCH * HD * 4);    // 1 MB
    float* g      = (float*)grab(NB * DCH * 4);
    float* z      = (float*)grab(BT * DCH * 4);
    float* y1     = (float*)grab(BT * DCH * 4);
    float* y2     = (float*)grab(BT * DCH * 4);
    float* weff1  = (float*)grab(DCH * DCH * 4);
    float* beff1  = (float*)grab(DCH * 4);
    float* weff2  = (float*)grab(DCH * DCH * 4);
    float* beff2  = (float*)grab(DCH * 4);
    float* adveff = (float*)grab(NA * DCH * 4);
    float* advbe  = (float*)grab(NA * 4);
    float* valeff = (float*)grab(DCH * 4);
    float* valbe  = (float*)grab(4);

    // --- CSR by destination (deterministic: per-node lists sorted) ---
    k_zero_i32<<<(NNODES + 255) / 256, 256, 0, stream>>>(deg, NNODES);
    k_count<<<(NEDGES + 255) / 256, 256, 0, stream>>>(dstv, deg, NEDGES);
    k_scan<<<1, 1024, 0, stream>>>(deg, rowptr, NNODES);
    k_copy_i32<<<(NNODES + 255) / 256, 256, 0, stream>>>(rowptr, cursor, NNODES);
    k_fill<<<(NEDGES + 255) / 256, 256, 0, stream>>>(dstv, cursor, eids, NEDGES);
    k_sortnode<<<(NNODES + 127) / 128, 128, 0, stream>>>(rowptr, eids, NNODES);

    // --- layer 1 (in=2): factored alpha tables + fused aggregate ---
    k_prep_p1<<<1, 64, 0, stream>>>(W1, q1, k1, Pq, Pk);
    k_agg1<<<NNODES, 128, 0, stream>>>(rowptr, eids, srcv, etype, x, W1, Pq, Pk, b1, h1);

    // --- layer 2: xw2 = h1 @ W2 (WMMA), aiaj = xw2 @ [q2|k2] (WMMA), fused aggregate ---
    k_w2t<<<(NREL * DCH * HD + 255) / 256, 256, 0, stream>>>(W2, W2T);
    k_wmma_gemm<DCH, 0><<<dim3(2048 / 64, NNODES / 16), 128, 0, stream>>>(
        h1, W2T, xw2, NNODES, 2048, DCH, DCH, 2048, nullptr);
    k_qkpadT<<<(16 * HD + 255) / 256, 256, 0, stream>>>(q2, k2, qkpT);
    k_wmma_gemm<HD, 0><<<dim3(1, (NNODES * NREL) / 16), 128, 0, stream>>>(
        xw2, qkpT, aiaj, NNODES * NREL, 16, HD, HD, 16, nullptr);
    k_agg2<<<NNODES, 128, 0, stream>>>(rowptr, eids, srcv, etype, aiaj, xw2, b2, h2);

    // --- pool + cosine embedding ---
    k_pool<<<NB, DCH, 0, stream>>>(h2, g);
    k_cosz<<<BT, DCH, 0, stream>>>(taus, cosw, cosb, g, z);

    // --- noisy effective weights ---
    k_noisy<<<(DCH * DCH + 255) / 256, 256, 0, stream>>>((const float*)d_in[14], (const float*)d_in[16], (const float*)d_in[18], weff1, DCH * DCH);
    k_noisy<<<1, 128, 0, stream>>>((const float*)d_in[15], (const float*)d_in[17], (const float*)d_in[19], beff1, DCH);
    k_noisy<<<(DCH * DCH + 255) / 256, 256, 0, stream>>>((const float*)d_in[20], (const float*)d_in[22], (const float*)d_in[24], weff2, DCH * DCH);
    k_noisy<<<1, 128, 0, stream>>>((const float*)d_in[21], (const float*)d_in[23], (const float*)d_in[25], beff2, DCH);
    k_noisy<<<(NA * DCH + 255) / 256, 256, 0, stream>>>((const float*)d_in[26], (const float*)d_in[28], (const float*)d_in[30], adveff, NA * DCH);
    k_noisy<<<1, 32, 0, stream>>>((const float*)d_in[27], (const float*)d_in[29], (const float*)d_in[31], advbe, NA);
    k_noisy<<<1, 128, 0, stream>>>((const float*)d_in[32], (const float*)d_in[34], (const float*)d_in[36], valeff, DCH);
    k_noisy<<<1, 32, 0, stream>>>((const float*)d_in[33], (const float*)d_in[35], (const float*)d_in[37], valbe, 1);

    // --- FF layers (WMMA, fused selu+bias); weff is already [out][in] = B^T layout ---
    k_wmma_gemm<DCH, 1><<<dim3(DCH / 64, BT / 16), 128, 0, stream>>>(
        z, weff1, y1, BT, DCH, DCH, DCH, DCH, beff1);
    k_wmma_gemm<DCH, 1><<<dim3(DCH / 64, BT / 16), 128, 0, stream>>>(
        y1, weff2, y2, BT, DCH, DCH, DCH, DCH, beff2);

    // --- dueling head + taus passthrough (tuple output) ---
    k_head<<<BT, 32, 0, stream>>>(y2, adveff, advbe, valeff, valbe, out);
    k_copy_f32<<<(BT + 255) / 256, 256, 0, stream>>>(taus, out + BT * NA, BT);
    (void)in_sizes; (void)n_in; (void)out_size; (void)ws_size;
}